// SelfAttention_77893526880607
// MI455X (gfx1250) — compile-verified
//
#include <hip/hip_runtime.h>
#include <hip/hip_bf16.h>

// Self-attention forward for B=2, S=2048, HID=1024, NH=16, HD=64 (all f32).
// Pipeline: qkv projections -> per-head QK^T logits -> row softmax (in place,
// score is output #1) -> score @ V (output #2, stored in [B,NH,S,HD] flat
// order to reproduce the reference's reshape-without-transpose).
// All matrix math uses V_WMMA_F32_16X16X4_F32 (exact f32, matches reference).
// All accesses to the 512MB score matrix are non-temporal (TH=NT) so the
// 192MB L2 keeps the reusable q/k/v/weight working set resident.

#define B_   2
#define S_   2048
#define HID_ 1024
#define NH_  16
#define HD_  64

typedef __attribute__((ext_vector_type(2))) float v2f;
typedef __attribute__((ext_vector_type(8))) float v8f;

// ---- f32 WMMA 16x16x4 fragment helpers -------------------------------------
// A (16x4, MxK): lane l<16 holds row M=l, K={k,k+1}; lane l>=16 holds row
// M=l-16, K={k+2,k+3}. B (4x16, KxN) is the mirror with N = lane&15.
// For an NT-major operand (row-major [rows, ld] with K contiguous) both A and
// B fragments use the same load pattern:
__device__ __forceinline__ v2f frag_nt(const float* __restrict__ base,
                                       int row0, int k, int ld) {
  const int lane = threadIdx.x & 31;
  const float* p = base + (size_t)(row0 + (lane & 15)) * ld
                        + (k + ((lane >> 4) << 1));
  return *(const v2f*)p;   // 8B aligned: k is a multiple of 4, ld of 64
}

// Same pattern, but streaming (non-temporal) — for the giant score matrix.
__device__ __forceinline__ v2f frag_nt_stream(const float* __restrict__ base,
                                              int row0, int k, int ld) {
  const int lane = threadIdx.x & 31;
  const float* p = base + (size_t)(row0 + (lane & 15)) * ld
                        + (k + ((lane >> 4) << 1));
  return __builtin_nontemporal_load((const v2f*)p);
}

__device__ __forceinline__ v8f wmma4(v2f a, v2f b, v8f c) {
  return __builtin_amdgcn_wmma_f32_16x16x4_f32(
      /*neg_a=*/false, a, /*neg_b=*/false, b,
      /*c_mod=*/(short)0, c, /*reuse_a=*/false, /*reuse_b=*/false);
}

__device__ __forceinline__ v8f vzero8() {
  v8f z = {0.f, 0.f, 0.f, 0.f, 0.f, 0.f, 0.f, 0.f};
  return z;
}

// ---- Kernel 1: Q/K/V projection y = x @ W^T, stored as [B,NH,S,HD] ---------
// Block = 256 threads = 8 waves arranged 2(M) x 4(N); block tile 64x256,
// wave tile 32x64 (2x4 accumulators of 16x16).
__global__ __launch_bounds__(256)
void qkv_proj_kernel(const float* __restrict__ x,
                     const float* __restrict__ wq,
                     const float* __restrict__ wk,
                     const float* __restrict__ wv,
                     float* __restrict__ qws,
                     float* __restrict__ kws,
                     float* __restrict__ vws) {
  const float* w;
  float* out;
  if (blockIdx.z == 0)      { w = wq; out = qws; }
  else if (blockIdx.z == 1) { w = wk; out = kws; }
  else                      { w = wv; out = vws; }

  const int wave = threadIdx.x >> 5;
  const int lane = threadIdx.x & 31;
  const int m0 = blockIdx.x * 64  + (wave >> 2) * 32;   // row in [0,4096)
  const int n0 = blockIdx.y * 256 + (wave & 3) * 64;    // col in [0,1024)

  v8f acc[2][4];
#pragma unroll
  for (int i = 0; i < 2; ++i)
#pragma unroll
    for (int t = 0; t < 4; ++t) acc[i][t] = vzero8();

  for (int k = 0; k < HID_; k += 4) {
    v2f a0 = frag_nt(x, m0,      k, HID_);
    v2f a1 = frag_nt(x, m0 + 16, k, HID_);
    v2f b[4];
#pragma unroll
    for (int t = 0; t < 4; ++t) b[t] = frag_nt(w, n0 + 16 * t, k, HID_);
#pragma unroll
    for (int t = 0; t < 4; ++t) {
      acc[0][t] = wmma4(a0, b[t], acc[0][t]);
      acc[1][t] = wmma4(a1, b[t], acc[1][t]);
    }
  }

  // C/D layout: vgpr r, lanes 0-15 -> (M=r, N=lane); lanes 16-31 -> (M=r+8).
  const int col_lo = lane & 15;
  const int half   = lane >> 4;
#pragma unroll
  for (int i = 0; i < 2; ++i) {
#pragma unroll
    for (int t = 0; t < 4; ++t) {
      const int n = n0 + 16 * t + col_lo;      // output feature j
      const int h = n >> 6, d = n & 63;
#pragma unroll
      for (int r = 0; r < 8; ++r) {
        const int m = m0 + 16 * i + r + 8 * half;  // token index b*S+s
        const int b = m >> 11, s = m & (S_ - 1);
        out[((size_t)(b * NH_ + h) * S_ + s) * HD_ + d] = acc[i][t][r];
      }
    }
  }
}

// ---- Kernel 2: per-head logits  att = (Q @ K^T) * 1/sqrt(HD) ---------------
// z = b*NH+h. Same NT tiling; K-dim = 64. Score stores are non-temporal.
__global__ __launch_bounds__(256)
void att_logits_kernel(const float* __restrict__ qws,
                       const float* __restrict__ kws,
                       float* __restrict__ score) {
  const int z = blockIdx.z;
  const float* Q = qws   + (size_t)z * S_ * HD_;
  const float* K = kws   + (size_t)z * S_ * HD_;
  float*       Y = score + (size_t)z * S_ * S_;

  const int wave = threadIdx.x >> 5;
  const int lane = threadIdx.x & 31;
  const int m0 = blockIdx.x * 64  + (wave >> 2) * 32;
  const int n0 = blockIdx.y * 256 + (wave & 3) * 64;

  v8f acc[2][4];
#pragma unroll
  for (int i = 0; i < 2; ++i)
#pragma unroll
    for (int t = 0; t < 4; ++t) acc[i][t] = vzero8();

  for (int k = 0; k < HD_; k += 4) {
    v2f a0 = frag_nt(Q, m0,      k, HD_);
    v2f a1 = frag_nt(Q, m0 + 16, k, HD_);
    v2f b[4];
#pragma unroll
    for (int t = 0; t < 4; ++t) b[t] = frag_nt(K, n0 + 16 * t, k, HD_);
#pragma unroll
    for (int t = 0; t < 4; ++t) {
      acc[0][t] = wmma4(a0, b[t], acc[0][t]);
      acc[1][t] = wmma4(a1, b[t], acc[1][t]);
    }
  }

  const float scale = 0.125f;   // 1/sqrt(64)
  const int col_lo = lane & 15;
  const int half   = lane >> 4;
#pragma unroll
  for (int i = 0; i < 2; ++i) {
#pragma unroll
    for (int t = 0; t < 4; ++t) {
      const int n = n0 + 16 * t + col_lo;
#pragma unroll
      for (int r = 0; r < 8; ++r) {
        const int m = m0 + 16 * i + r + 8 * half;
        __builtin_nontemporal_store(acc[i][t][r] * scale,
                                    &Y[(size_t)m * S_ + n]);
      }
    }
  }
}

// ---- Kernel 3: row softmax in place (one block per 2048-elem row) ----------
// Streaming row: NT loads and stores.
__global__ __launch_bounds__(256)
void softmax_kernel(float* __restrict__ score) {
  __shared__ float red[256];
  float* p = score + (size_t)blockIdx.x * S_;
  const int t = threadIdx.x;

  float v[8];
  float m = -3.402823466e38f;
#pragma unroll
  for (int i = 0; i < 8; ++i) {
    v[i] = __builtin_nontemporal_load(p + t + 256 * i);
    m = fmaxf(m, v[i]);
  }
  red[t] = m;
  __syncthreads();
  for (int off = 128; off > 0; off >>= 1) {
    if (t < off) red[t] = fmaxf(red[t], red[t + off]);
    __syncthreads();
  }
  m = red[0];
  __syncthreads();

  float s = 0.f;
#pragma unroll
  for (int i = 0; i < 8; ++i) {
    v[i] = __expf(v[i] - m);
    s += v[i];
  }
  red[t] = s;
  __syncthreads();
  for (int off = 128; off > 0; off >>= 1) {
    if (t < off) red[t] += red[t + off];
    __syncthreads();
  }
  const float inv = 1.0f / red[0];
#pragma unroll
  for (int i = 0; i < 8; ++i)
    __builtin_nontemporal_store(v[i] * inv, p + t + 256 * i);
}

// ---- Kernel 4: attv = score @ V per head (NN GEMM, N = 64) -----------------
// Block = 8 waves stacked in M: block tile 256x64, wave tile 32x64, K = 2048.
// Score (A) fragment loads are non-temporal; V stays hot in L2.
__global__ __launch_bounds__(256)
void attv_kernel(const float* __restrict__ score,
                 const float* __restrict__ vws,
                 float* __restrict__ out) {
  const int z = blockIdx.z;
  const float* A = score + (size_t)z * S_ * S_;
  const float* V = vws   + (size_t)z * S_ * HD_;
  float*       Y = out   + (size_t)z * S_ * HD_;

  const int wave = threadIdx.x >> 5;
  const int lane = threadIdx.x & 31;
  const int m0 = blockIdx.x * 256 + wave * 32;
  const int col_lo = lane & 15;
  const int half   = lane >> 4;

  v8f acc[2][4];
#pragma unroll
  for (int i = 0; i < 2; ++i)
#pragma unroll
    for (int t = 0; t < 4; ++t) acc[i][t] = vzero8();

  for (int k = 0; k < S_; k += 4) {
    v2f a0 = frag_nt_stream(A, m0,      k, S_);
    v2f a1 = frag_nt_stream(A, m0 + 16, k, S_);
    v2f b[4];
#pragma unroll
    for (int t = 0; t < 4; ++t) {
      // B fragment (NN): lane n holds V[k+2*half, n] and V[k+2*half+1, n]
      const float* bp = V + (size_t)(k + 2 * half) * HD_ + 16 * t + col_lo;
      v2f bb;
      bb.x = bp[0];
      bb.y = bp[HD_];
      b[t] = bb;
    }
#pragma unroll
    for (int t = 0; t < 4; ++t) {
      acc[0][t] = wmma4(a0, b[t], acc[0][t]);
      acc[1][t] = wmma4(a1, b[t], acc[1][t]);
    }
  }

#pragma unroll
  for (int i = 0; i < 2; ++i) {
#pragma unroll
    for (int t = 0; t < 4; ++t) {
      const int n = 16 * t + col_lo;
#pragma unroll
      for (int r = 0; r < 8; ++r) {
        const int m = m0 + 16 * i + r + 8 * half;
        Y[(size_t)m * HD_ + n] = acc[i][t][r];
      }
    }
  }
}

extern "C" void kernel_launch(void* const* d_in, const int* in_sizes, int n_in,
                              void* d_out, int out_size, void* d_ws, size_t ws_size,
                              hipStream_t stream) {
  const float* x  = (const float*)d_in[0];
  const float* wq = (const float*)d_in[1];
  const float* wk = (const float*)d_in[2];
  const float* wv = (const float*)d_in[3];

  float* score = (float*)d_out;                                   // [B,NH,S,S]
  float* attv  = score + (size_t)B_ * NH_ * S_ * S_;              // [B,NH,S,HD] flat
  float* qws = (float*)d_ws;                                      // 16 MB each
  float* kws = qws + (size_t)B_ * NH_ * S_ * HD_;
  float* vws = kws + (size_t)B_ * NH_ * S_ * HD_;

  dim3 blk(256);
  qkv_proj_kernel<<<dim3((B_ * S_) / 64, HID_ / 256, 3), blk, 0, stream>>>(
      x, wq, wk, wv, qws, kws, vws);
  att_logits_kernel<<<dim3(S_ / 64, S_ / 256, B_ * NH_), blk, 0, stream>>>(
      qws, kws, score);
  softmax_kernel<<<dim3((unsigned)((size_t)B_ * NH_ * S_)), blk, 0, stream>>>(score);
  attv_kernel<<<dim3(S_ / 256, 1, B_ * NH_), blk, 0, stream>>>(score, vws, attv);
}